// LinearDeformableCrossAttn_5231270166876
// MI455X (gfx1250) — compile-verified
//
#include <hip/hip_runtime.h>
#include <hip/hip_bf16.h>
#include <math.h>

#define NHEADS   8
#define NPOINTS  4
#define C_DIM    256
#define HW       4096
#define B_DIM    8
#define HK       64
#define WK       64
#define D_HEAD   32

typedef __attribute__((ext_vector_type(2))) float v2f;
typedef __attribute__((ext_vector_type(8))) float v8f;

// ---------------------------------------------------------------------------
// Kernel A: offset GEMM (q @ w_off.T + b_off) + softmax + sample coordinates
// Block: 128 threads, 32 pixels. Thread owns 24 consecutive j's = 2 full heads.
// ---------------------------------------------------------------------------
#define PIX_PER_BLK 32
#define J_PER_THREAD 24

__global__ __launch_bounds__(128) void deform_offsets_kernel(
    const float* __restrict__ query, const float* __restrict__ w_off,
    const float* __restrict__ b_off, float4* __restrict__ params)
{
  __shared__ float lds_q[C_DIM * (PIX_PER_BLK + 1)];
  const int b      = blockIdx.x / (HW / PIX_PER_BLK);
  const int pixel0 = (blockIdx.x % (HW / PIX_PER_BLK)) * PIX_PER_BLK;
  const int tid    = threadIdx.x;

  // Stage query tile [c][p] (coalesced over p, padded stride 33 in LDS)
  const float* qb = query + (size_t)b * C_DIM * HW + pixel0;
  for (int i = tid; i < C_DIM * PIX_PER_BLK; i += 128) {
    int c = i >> 5;
    int p = i & 31;
    lds_q[c * (PIX_PER_BLK + 1) + p] = qb[(size_t)c * HW + p];
  }
  __syncthreads();

  const int p     = tid & 31;       // pixel within tile (lane index)
  const int jset  = tid >> 5;       // wave-uniform
  const int jbase = jset * J_PER_THREAD;

  float acc[J_PER_THREAD];
  #pragma unroll
  for (int jj = 0; jj < J_PER_THREAD; ++jj) acc[jj] = b_off[jbase + jj];

  for (int c = 0; c < C_DIM; ++c) {
    float qv = lds_q[c * (PIX_PER_BLK + 1) + p];
    #pragma unroll
    for (int jj = 0; jj < J_PER_THREAD; ++jj)   // w_off index wave-uniform -> s_load
      acc[jj] = fmaf(qv, w_off[(jbase + jj) * C_DIM + c], acc[jj]);
  }

  const int pixel = pixel0 + p;
  const float px = (float)(pixel & (WK - 1));
  const float py = (float)(pixel >> 6);

  #pragma unroll
  for (int hh = 0; hh < 2; ++hh) {
    const int head = jset * 2 + hh;
    const int off  = hh * 12;  // j = head*12 + pt*3 + comp
    float s0 = acc[off + 2],  s1 = acc[off + 5];
    float s2 = acc[off + 8],  s3 = acc[off + 11];
    float m  = fmaxf(fmaxf(s0, s1), fmaxf(s2, s3));
    float e0 = __expf(s0 - m), e1 = __expf(s1 - m);
    float e2 = __expf(s2 - m), e3 = __expf(s3 - m);
    float inv = 1.0f / (e0 + e1 + e2 + e3);
    float wv[4] = {e0 * inv, e1 * inv, e2 * inv, e3 * inv};

    float4* outp = params + (((size_t)b * HW + pixel) * NHEADS + head) * NPOINTS;
    #pragma unroll
    for (int pt = 0; pt < NPOINTS; ++pt) {
      // xg = (gx + 0.1*ox + 1)*0.5*(Wk-1) = x + 3.15*ox  (since gx+1 = 2x/63)
      float xg = px + acc[off + pt * 3 + 0] * 3.15f;
      float yg = py + acc[off + pt * 3 + 1] * 3.15f;
      outp[pt] = make_float4(xg, yg, wv[pt], 0.0f);
    }
  }
}

// ---------------------------------------------------------------------------
// Kernel B: bilinear sampling + softmax-weighted sum over points.
// One thread per (b, pixel, head); 32-channel slab written contiguously.
// ---------------------------------------------------------------------------
__global__ __launch_bounds__(256) void deform_sample_kernel(
    const float* __restrict__ key, const float4* __restrict__ params,
    float* __restrict__ feat)
{
  const int tid = blockIdx.x * 256 + threadIdx.x;
  const int h   = tid & 7;
  const int pix = (tid >> 3) & (HW - 1);
  const int b   = tid >> 15;

  const float* kbase = key + ((size_t)b * C_DIM + h * D_HEAD) * (HK * WK);
  float acc[D_HEAD];
  #pragma unroll
  for (int d = 0; d < D_HEAD; ++d) acc[d] = 0.0f;

  const float4* prm = params + (((size_t)b * HW + pix) * NHEADS + h) * NPOINTS;

  #pragma unroll
  for (int pt = 0; pt < NPOINTS; ++pt) {
    float4 v = prm[pt];
    float xg = v.x, yg = v.y, w = v.z;
    float x0f = floorf(xg), y0f = floorf(yg);
    float fx = xg - x0f, fy = yg - y0f;
    int x0 = (int)x0f, y0 = (int)y0f;
    int x1 = x0 + 1,   y1 = y0 + 1;
    float vx0 = (x0 >= 0 && x0 <= WK - 1) ? 1.0f : 0.0f;
    float vx1 = (x1 >= 0 && x1 <= WK - 1) ? 1.0f : 0.0f;
    float vy0 = (y0 >= 0 && y0 <= HK - 1) ? 1.0f : 0.0f;
    float vy1 = (y1 >= 0 && y1 <= HK - 1) ? 1.0f : 0.0f;
    int cx0 = min(max(x0, 0), WK - 1), cx1 = min(max(x1, 0), WK - 1);
    int cy0 = min(max(y0, 0), HK - 1), cy1 = min(max(y1, 0), HK - 1);
    float w00 = (1.0f - fx) * (1.0f - fy) * vx0 * vy0 * w;
    float w10 = fx * (1.0f - fy) * vx1 * vy0 * w;
    float w01 = (1.0f - fx) * fy * vx0 * vy1 * w;
    float w11 = fx * fy * vx1 * vy1 * w;
    int i00 = cy0 * WK + cx0, i10 = cy0 * WK + cx1;
    int i01 = cy1 * WK + cx0, i11 = cy1 * WK + cx1;
    #pragma unroll
    for (int d = 0; d < D_HEAD; ++d) {
      const float* kd = kbase + (size_t)d * (HK * WK);
      acc[d] += w00 * kd[i00] + w10 * kd[i10] + w01 * kd[i01] + w11 * kd[i11];
    }
  }

  float* fp = feat + ((size_t)b * HW + pix) * C_DIM + h * D_HEAD;
  #pragma unroll
  for (int d = 0; d < D_HEAD; d += 4)
    *(float4*)(fp + d) = make_float4(acc[d], acc[d + 1], acc[d + 2], acc[d + 3]);
}

// ---------------------------------------------------------------------------
// Kernel C: projection GEMM out = feat[32768,256] @ w_proj.T + b_proj
// fp32 WMMA 16x16x4; each wave computes a 16x64 tile (4 accumulators).
// ---------------------------------------------------------------------------
__global__ __launch_bounds__(256) void proj_wmma_kernel(
    const float* __restrict__ feat, const float* __restrict__ w_proj,
    const float* __restrict__ b_proj, float* __restrict__ out)
{
  const int lane  = threadIdx.x & 31;
  const int wid   = (blockIdx.x * 256 + threadIdx.x) >> 5;
  const int tileN = wid & 3;       // 4 groups of 64 columns
  const int tileM = wid >> 2;      // 2048 row-tiles of 16
  const int m0 = tileM * 16;
  const int n0 = tileN * 64;
  const int l16 = lane & 15;
  const int lh  = lane >> 4;

  v8f acc[4];
  #pragma unroll
  for (int t = 0; t < 4; ++t) {
    float bias = b_proj[n0 + t * 16 + l16];   // D(m,n): n = lane&15 for all 8 regs
    #pragma unroll
    for (int r = 0; r < 8; ++r) acc[t][r] = bias;
  }

  // A frag: lane holds A[m = lane&15, k0 + (lane>>4)*2 + {0,1}]  (float2)
  const float* aptr  = feat   + (size_t)(m0 + l16) * C_DIM + lh * 2;
  // B frag: B(k,n) = w_proj[n,k]; lane holds n = lane&15, k = k0 + (lane>>4)*2 + {0,1}
  const float* bptr0 = w_proj + (size_t)(n0 + l16) * C_DIM + lh * 2;

  #pragma unroll 4
  for (int k0 = 0; k0 < C_DIM; k0 += 4) {
    v2f a = *(const v2f*)(aptr + k0);
    #pragma unroll
    for (int t = 0; t < 4; ++t) {
      v2f bm = *(const v2f*)(bptr0 + (size_t)t * 16 * C_DIM + k0);
      acc[t] = __builtin_amdgcn_wmma_f32_16x16x4_f32(
          false, a, false, bm, (short)0, acc[t], false, false);
    }
  }

  #pragma unroll
  for (int t = 0; t < 4; ++t) {
    float* op = out + (size_t)(m0 + lh * 8) * C_DIM + n0 + t * 16 + l16;
    #pragma unroll
    for (int r = 0; r < 8; ++r)
      op[(size_t)r * C_DIM] = acc[t][r];   // row m = m0 + lh*8 + r
  }
}

// ---------------------------------------------------------------------------
extern "C" void kernel_launch(void* const* d_in, const int* in_sizes, int n_in,
                              void* d_out, int out_size, void* d_ws, size_t ws_size,
                              hipStream_t stream) {
  (void)in_sizes; (void)n_in; (void)out_size; (void)ws_size;
  const float* query  = (const float*)d_in[0];
  const float* key    = (const float*)d_in[1];
  const float* w_off  = (const float*)d_in[2];
  const float* b_off  = (const float*)d_in[3];
  const float* w_proj = (const float*)d_in[4];
  const float* b_proj = (const float*)d_in[5];

  // Workspace layout: params (float4 per (b,pix,head,pt)) then feat [B,HW,C]
  float4* params = (float4*)d_ws;
  float*  feat   = (float*)d_ws + (size_t)B_DIM * HW * NHEADS * NPOINTS * 4;

  deform_offsets_kernel<<<B_DIM * (HW / PIX_PER_BLK), 128, 0, stream>>>(
      query, w_off, b_off, params);
  deform_sample_kernel<<<(B_DIM * HW * NHEADS) / 256, 256, 0, stream>>>(
      key, params, feat);
  proj_wmma_kernel<<<((B_DIM * HW / 16) * 4 * 32) / 256, 256, 0, stream>>>(
      feat, w_proj, b_proj, (float*)d_out);
}